// MultiBoxLoss_39247411151114
// MI455X (gfx1250) — compile-verified
//
#include <hip/hip_runtime.h>
#include <cstdint>
#include <cstddef>

#define TILE 256

typedef float v2f __attribute__((ext_vector_type(2)));
typedef float v8f __attribute__((ext_vector_type(8)));

// ---------------------------------------------------------------------------
// Wave-level sum of 32 lane values using V_WMMA_F32_16X16X4_F32 with a ones
// A-matrix: D[i][j] = sum_k A[i][k]*B[k][j] = colsum_j(B). With B VGPR0 = data
// and VGPR1 = 0, lane L of D's first VGPR holds v[L%16] + v[(L%16)+16]
// (all D rows identical), so 4 shfl_xor steps finish the reduction.
// Requires EXEC == all ones: call only from uniform control flow, full waves.
// ---------------------------------------------------------------------------
static __device__ __forceinline__ float wave_sum32(float v) {
#if defined(__gfx1250__) && __has_builtin(__builtin_amdgcn_wmma_f32_16x16x4_f32)
  v2f a = {1.0f, 1.0f};       // A = ones(16x4)
  v2f bb = {v, 0.0f};         // B = [data rows ; zero rows]
  v8f c = {0.f, 0.f, 0.f, 0.f, 0.f, 0.f, 0.f, 0.f};
  v8f d = __builtin_amdgcn_wmma_f32_16x16x4_f32(false, a, false, bb,
                                                (short)0, c, false, false);
  float s = d[0];             // pair-sum v[j] + v[j+16]
  s += __shfl_xor(s, 1, 32);
  s += __shfl_xor(s, 2, 32);
  s += __shfl_xor(s, 4, 32);
  s += __shfl_xor(s, 8, 32);
  return s;
#else
  for (int off = 16; off; off >>= 1) v += __shfl_xor(v, off, 32);
  return v;
#endif
}

// Block (256-thread / 8-wave) sum; valid result in thread 0.
static __device__ __forceinline__ float block_sum8(float v, float* red, int t) {
  float w = wave_sum32(v);
  int wid = t >> 5, lane = t & 31;
  if (lane == 0) red[wid] = w;
  __syncthreads();
  float s = 0.f;
  if (t == 0) {
#pragma unroll
    for (int i = 0; i < 8; ++i) s += red[i];
  }
  __syncthreads();  // allow red reuse
  return s;
}

// ---------------------------------------------------------------------------
// Async global->LDS staging (gfx1250). Guarded: falls back to plain loads.
// Builtin signature (from compiler diagnostic): param0 = int4* in AS1 (global),
// param1 = LDS destination, then imm offset, imm cpol.
// ---------------------------------------------------------------------------
#if defined(__gfx1250__) && __has_builtin(__builtin_amdgcn_global_load_async_to_lds_b128) && __has_builtin(__builtin_amdgcn_s_wait_asynccnt)
#define HAVE_ASYNC_LDS 1
typedef int v4i_t __attribute__((ext_vector_type(4)));
typedef __attribute__((address_space(1))) v4i_t* gv4_t;
typedef __attribute__((address_space(3))) v4i_t* lv4_t;
#endif

// ---------------------------------------------------------------------------
// K0: zero the small workspace pieces
// ---------------------------------------------------------------------------
__global__ void mbl_init_ws(unsigned long long* gt_max, float* pos_cnt,
                            float* neg_sum, float* scalars, int nGt, int B) {
  int i = blockIdx.x * blockDim.x + threadIdx.x;
  if (i < nGt) gt_max[i] = 0ull;
  if (i < B) { pos_cnt[i] = 0.f; neg_sum[i] = 0.f; }
  if (i < 4) scalars[i] = 0.f;
}

// ---------------------------------------------------------------------------
// K1: per-prior best gt (first-max over m) + per-gt best prior via packed
//     u64 atomicMax (iou_bits<<32 | ~p  => lowest p wins ties).
// ---------------------------------------------------------------------------
__global__ __launch_bounds__(TILE)
void mbl_match(const float4* __restrict__ dbox, const float4* __restrict__ gbox,
               unsigned long long* __restrict__ packed_prior,
               unsigned long long* __restrict__ gt_max, int P, int M) {
  int b = blockIdx.y;
  int p = blockIdx.x * TILE + threadIdx.x;
  int t = threadIdx.x;
  __shared__ float4 g_s[64];
  __shared__ float garea[64];
  __shared__ unsigned long long smax[64];
  if (t < M) {
    float4 g = gbox[b * M + t];
    g_s[t] = g;
    garea[t] = (g.z - g.x) * (g.w - g.y);
    smax[t] = 0ull;
  }
  __syncthreads();
  bool valid = p < P;
  float4 d = {0.f, 0.f, 0.f, 0.f};
  float darea = 0.f;
  if (valid) { d = dbox[p]; darea = (d.z - d.x) * (d.w - d.y); }
  float best = -1.f;
  int bestm = 0;
  unsigned key_lo = ~(unsigned)p;
  for (int m = 0; m < M; ++m) {
    if (valid) {
      float4 g = g_s[m];
      float lx = fmaxf(d.x, g.x), ly = fmaxf(d.y, g.y);
      float rx = fminf(d.z, g.z), ry = fminf(d.w, g.w);
      float w = fmaxf(rx - lx, 0.f), h = fmaxf(ry - ly, 0.f);
      float inter = w * h;
      float iou = inter / (darea + garea[m] - inter);
      if (iou > best) { best = iou; bestm = m; }   // strict > keeps first max
      unsigned long long key =
          ((unsigned long long)__float_as_uint(iou) << 32) | (unsigned long long)key_lo;
      atomicMax(&smax[m], key);
    }
  }
  __syncthreads();
  if (t < M) atomicMax(&gt_max[(size_t)b * M + t], smax[t]);
  if (valid)
    packed_prior[(size_t)b * P + p] =
        ((unsigned long long)__float_as_uint(best) << 32) | (unsigned)bestm;
}

// ---------------------------------------------------------------------------
// K2: force each gt's best prior to (iou=1.0, idx=m); sequential m => last wins
// ---------------------------------------------------------------------------
__global__ void mbl_override(const unsigned long long* __restrict__ gt_max,
                             unsigned long long* __restrict__ packed_prior,
                             int B, int M, int P) {
  int b = blockIdx.x * blockDim.x + threadIdx.x;
  if (b >= B) return;
  for (int m = 0; m < M; ++m) {
    unsigned long long key = gt_max[(size_t)b * M + m];
    unsigned p = ~(unsigned)(key & 0xFFFFFFFFull);
    if (p < (unsigned)P)
      packed_prior[(size_t)b * P + p] =
          (((unsigned long long)0x3F800000u) << 32) | (unsigned)m;  // iou = 1.0f
  }
}

// ---------------------------------------------------------------------------
// K3: fused per-prior CE (log-softmax over 21 classes, tile staged in LDS via
//     async copy), positive smooth-L1 loc loss, pos counts; WMMA reductions.
// ---------------------------------------------------------------------------
__global__ __launch_bounds__(TILE)
void mbl_conf(const float* __restrict__ conf, const float4* __restrict__ locp,
              const float4* __restrict__ dbox, const float4* __restrict__ gbox,
              const int* __restrict__ glab,
              const unsigned long long* __restrict__ packed_prior,
              float* __restrict__ ce_neg, float* __restrict__ pos_cnt,
              float* __restrict__ scalars, int P, int M) {
  const int C = 21;
  int b = blockIdx.y;
  int p0 = blockIdx.x * TILE;
  int t = threadIdx.x;
  int p = p0 + t;

  __shared__ float4 g_s[64];
  __shared__ int lab_s[64];
  __shared__ __align__(16) float tile[TILE * 21];
  __shared__ float red[8];

  if (t < M) { g_s[t] = gbox[b * M + t]; lab_s[t] = glab[b * M + t]; }

  int nprior = min(TILE, P - p0);
  int ndw = nprior * C;
  const float* src = conf + ((size_t)b * P + (size_t)p0) * C;

#ifdef HAVE_ASYNC_LDS
  if ((((size_t)(const void*)src) & 15) == 0) {
    gv4_t gs = (gv4_t)(const void*)src;
    lv4_t ls = (lv4_t)(void*)&tile[0];
    int nvec = ndw >> 2;
    for (int i = t; i < nvec; i += TILE)
      __builtin_amdgcn_global_load_async_to_lds_b128(gs + i, ls + i, 0, 0);
    for (int i = (nvec << 2) + t; i < ndw; i += TILE) tile[i] = src[i];
    __builtin_amdgcn_s_wait_asynccnt(0);
  } else
#endif
  {
    for (int i = t; i < ndw; i += TILE) tile[i] = src[i];
  }
  __syncthreads();

  bool valid = p < P;
  float l1 = 0.f, ce = 0.f, isPos = 0.f;
  if (valid) {
    unsigned long long pk = packed_prior[(size_t)b * P + p];
    float iou = __uint_as_float((unsigned)(pk >> 32));
    int m = (int)(pk & 0xFFFFFFFFull);
    int lbl = (iou < 0.5f) ? 0 : lab_s[m];

    const float* x = &tile[t * C];
    float mx = x[0];
#pragma unroll
    for (int c = 1; c < C; ++c) mx = fmaxf(mx, x[c]);
    float se = 0.f;
#pragma unroll
    for (int c = 0; c < C; ++c) se += expf(x[c] - mx);
    float lse = mx + logf(se);
    ce = lse - x[lbl];   // >= 0

    if (lbl > 0) {
      isPos = 1.f;
      float4 d = dbox[p];
      float4 g = g_s[m];
      float dcx = 0.5f * (d.x + d.z), dcy = 0.5f * (d.y + d.w);
      float dw = d.z - d.x, dh = d.w - d.y;
      float gcx = 0.5f * (g.x + g.z), gcy = 0.5f * (g.y + g.w);
      float gw = g.z - g.x, gh = g.w - g.y;
      float e0 = (gcx - dcx) / (0.1f * dw);
      float e1 = (gcy - dcy) / (0.1f * dh);
      float e2 = logf(gw / dw) / 0.2f;
      float e3 = logf(gh / dh) / 0.2f;
      float4 lp = locp[(size_t)b * P + p];
      float df;
      df = fabsf(lp.x - e0); l1 += (df < 1.f) ? 0.5f * df * df : df - 0.5f;
      df = fabsf(lp.y - e1); l1 += (df < 1.f) ? 0.5f * df * df : df - 0.5f;
      df = fabsf(lp.z - e2); l1 += (df < 1.f) ? 0.5f * df * df : df - 0.5f;
      df = fabsf(lp.w - e3); l1 += (df < 1.f) ? 0.5f * df * df : df - 0.5f;
      ce_neg[(size_t)b * P + p] = 0.f;
    } else {
      ce_neg[(size_t)b * P + p] = ce;
    }
  }

  float cepos = isPos * ce;
  float tl = block_sum8(l1, red, t);
  float tc = block_sum8(cepos, red, t);
  float tp = block_sum8(isPos, red, t);
  if (t == 0) {
    if (tl != 0.f) atomicAdd(&scalars[0], tl);
    if (tc != 0.f) atomicAdd(&scalars[1], tc);
    if (tp != 0.f) atomicAdd(&pos_cnt[b], tp);
  }
}

// ---------------------------------------------------------------------------
// K4: per-batch exact top-k sum of ce_neg via 4-pass MSD radix select on the
//     (order-preserving, non-negative) float bit patterns.
//     neg_sum[b] = sum_{x > t} x + (k - count_gt) * t,  t = k-th largest.
// ---------------------------------------------------------------------------
__global__ __launch_bounds__(TILE)
void mbl_select(const float* __restrict__ ce_neg, const float* __restrict__ pos_cnt,
                float* __restrict__ neg_sum, int P) {
  int b = blockIdx.x;
  const float* row = ce_neg + (size_t)b * P;
  int k = 3 * (int)(pos_cnt[b] + 0.5f);
  if (k > P) k = P;
  if (k <= 0) {
    if (threadIdx.x == 0) neg_sum[b] = 0.f;
    return;  // uniform for whole block
  }

  __shared__ unsigned hist[256];
  __shared__ unsigned sh_prefix;
  __shared__ int sh_krem;
  __shared__ float red[8];

  unsigned prefix = 0;
  int krem = k;
  for (int pass = 0; pass < 4; ++pass) {
    int shift = 24 - 8 * pass;
    unsigned himask = (pass == 0) ? 0u : (0xFFFFFFFFu << (shift + 8));
    for (int i = threadIdx.x; i < 256; i += blockDim.x) hist[i] = 0u;
    __syncthreads();
    for (int p = threadIdx.x; p < P; p += blockDim.x) {
      unsigned bits = __float_as_uint(row[p]);
      if ((bits & himask) == prefix) atomicAdd(&hist[(bits >> shift) & 0xFF], 1u);
    }
    __syncthreads();
    if (threadIdx.x == 0) {
      int cum = 0, kr = krem;
      unsigned byte = 0;
      for (int v = 255; v >= 0; --v) {
        int h = (int)hist[v];
        if (cum + h >= kr) { byte = (unsigned)v; kr -= cum; break; }
        cum += h;
      }
      sh_prefix = prefix | (byte << shift);
      sh_krem = kr;
    }
    __syncthreads();
    prefix = sh_prefix;
    krem = sh_krem;
    __syncthreads();
  }

  float tval = __uint_as_float(prefix);
  float s = 0.f;
  for (int p = threadIdx.x; p < P; p += blockDim.x) {
    unsigned bits = __float_as_uint(row[p]);
    if (bits > prefix) s += __uint_as_float(bits);
  }
  float tot = block_sum8(s, red, threadIdx.x);
  if (threadIdx.x == 0) neg_sum[b] = tot + (float)krem * tval;
}

// ---------------------------------------------------------------------------
// K5: final scalar
// ---------------------------------------------------------------------------
__global__ void mbl_finalize(const float* __restrict__ pos_cnt,
                             const float* __restrict__ neg_sum,
                             const float* __restrict__ scalars,
                             float* __restrict__ out, int B) {
  if (threadIdx.x == 0 && blockIdx.x == 0) {
    float tp = 0.f, ns = 0.f;
    for (int i = 0; i < B; ++i) { tp += pos_cnt[i]; ns += neg_sum[i]; }
    float np = fmaxf(tp, 1.f);
    out[0] = (scalars[0] + scalars[1] + ns) / np;
  }
}

// ---------------------------------------------------------------------------
extern "C" void kernel_launch(void* const* d_in, const int* in_sizes, int n_in,
                              void* d_out, int out_size, void* d_ws, size_t ws_size,
                              hipStream_t stream) {
  (void)n_in; (void)out_size; (void)ws_size;
  const float* loc  = (const float*)d_in[0];   // (B,P,4)
  const float* conf = (const float*)d_in[1];   // (B,P,C)
  const float* gtb  = (const float*)d_in[2];   // (B,M,4)
  const int*   gtl  = (const int*)d_in[3];     // (B,M)
  const float* dbx  = (const float*)d_in[4];   // (P,4)

  int P = in_sizes[4] / 4;
  int B = in_sizes[0] / (P * 4);
  int M = in_sizes[3] / B;

  char* ws = (char*)d_ws;
  size_t off = 0;
  auto take = [&](size_t bytes) {
    char* q = ws + off;
    off += (bytes + 255) & ~(size_t)255;
    return q;
  };
  unsigned long long* packed = (unsigned long long*)take((size_t)B * P * 8);
  float* ce_neg              = (float*)take((size_t)B * P * 4);
  unsigned long long* gt_max = (unsigned long long*)take((size_t)B * M * 8);
  float* pos_cnt             = (float*)take((size_t)B * 4);
  float* neg_sum             = (float*)take((size_t)B * 4);
  float* scalars             = (float*)take(16);

  int nGt = B * M;
  mbl_init_ws<<<dim3((nGt + 255) / 256), dim3(256), 0, stream>>>(
      gt_max, pos_cnt, neg_sum, scalars, nGt, B);

  dim3 gBP((P + TILE - 1) / TILE, B);
  mbl_match<<<gBP, dim3(TILE), 0, stream>>>(
      (const float4*)dbx, (const float4*)gtb, packed, gt_max, P, M);

  mbl_override<<<dim3((B + 63) / 64), dim3(64), 0, stream>>>(gt_max, packed, B, M, P);

  mbl_conf<<<gBP, dim3(TILE), 0, stream>>>(
      conf, (const float4*)loc, (const float4*)dbx, (const float4*)gtb, gtl,
      packed, ce_neg, pos_cnt, scalars, P, M);

  mbl_select<<<dim3(B), dim3(TILE), 0, stream>>>(ce_neg, pos_cnt, neg_sum, P);

  mbl_finalize<<<dim3(1), dim3(1), 0, stream>>>(pos_cnt, neg_sum, scalars,
                                                (float*)d_out, B);
}